// FaultPredictionModel_62483184222584
// MI455X (gfx1250) — compile-verified
//
#include <hip/hip_runtime.h>

typedef __attribute__((ext_vector_type(2))) float v2f;
typedef __attribute__((ext_vector_type(8))) float v8f;

#define LENG 2048
#define NSEN 32
#define NBAT 32
#define NFIL 64
#define CLSIN 2176

// ---------------- softmax over 32 sensor weights (one wave) ----------------
__global__ __launch_bounds__(32) void softmax32_kernel(const float* __restrict__ w_in,
                                                       float* __restrict__ w_out) {
    int t = threadIdx.x;
    float w = w_in[t];
    float m = w;
    for (int off = 16; off > 0; off >>= 1) m = fmaxf(m, __shfl_xor(m, off, 32));
    float e = expf(w - m);
    float s = e;
    for (int off = 16; off > 0; off >>= 1) s += __shfl_xor(s, off, 32);
    w_out[t] = e / s;
}

// ---- per-(b,s): weighted load + kd stats + conv/relu/maxpool via WMMA f32 ----
__global__ __launch_bounds__(256) void conv_kernel(const float* __restrict__ x,
                                                   const float* __restrict__ conv_w,
                                                   const float* __restrict__ conv_b,
                                                   const float* __restrict__ sw,
                                                   float* __restrict__ fused) {
    __shared__ float  xs[LENG + 4];        // halo-padded: xs[l+1] = xw[l], extra tail pad
    __shared__ float4 red[256];            // sum, sumsq, max, min
    __shared__ float  wavemax[8][NFIL];

    const int tid = threadIdx.x;
    const int b   = blockIdx.x >> 5;
    const int s   = blockIdx.x & 31;
    const float ws_s = sw[s];

    // ---- load + per-thread stats (x layout [B, L, S] -> stride S in l) ----
    float sum = 0.f, sumsq = 0.f;
    float vmax = -__builtin_inff(), vmin = __builtin_inff();
    const float* xb = x + (size_t)b * LENG * NSEN + s;
    #pragma unroll
    for (int j = 0; j < LENG / 256; ++j) {
        int l = tid + j * 256;
        float v = xb[(size_t)l * NSEN] * ws_s;
        xs[l + 1] = v;
        sum += v; sumsq += v * v;
        vmax = fmaxf(vmax, v); vmin = fminf(vmin, v);
    }
    if (tid == 0) { xs[0] = 0.f; xs[LENG + 1] = 0.f; xs[LENG + 2] = 0.f; xs[LENG + 3] = 0.f; }
    red[tid] = make_float4(sum, sumsq, vmax, vmin);
    __syncthreads();
    for (int stride = 128; stride > 0; stride >>= 1) {
        if (tid < stride) {
            float4 a = red[tid], c = red[tid + stride];
            red[tid] = make_float4(a.x + c.x, a.y + c.y, fmaxf(a.z, c.z), fminf(a.w, c.w));
        }
        __syncthreads();
    }

    // ---- conv as GEMM: A[16x4] = shifted taps, B[4x16] = filters, K=4 (tap3 = 0) ----
    // wave is uniform within the wave; pin it to an SGPR so loop control is scalar.
    const int wave  = __builtin_amdgcn_readfirstlane(tid >> 5);
    const int lane  = tid & 31;
    const int m16   = lane & 15;   // row of A / col of B,D
    const int khalf = lane >> 4;   // selects K pair {0,1} or {2,3}

    v2f bmat[4];
    v8f cbias[4];                  // bias as WMMA C operand: C[m][n] = conv_b[n]
    #pragma unroll
    for (int nt = 0; nt < 4; ++nt) {
        int n = nt * 16 + m16;
        if (khalf == 0) { bmat[nt].x = conv_w[n * 3 + 0]; bmat[nt].y = conv_w[n * 3 + 1]; }
        else            { bmat[nt].x = conv_w[n * 3 + 2]; bmat[nt].y = 0.f; }
        float bb = conv_b[n];
        #pragma unroll
        for (int r = 0; r < 8; ++r) cbias[nt][r] = bb;
    }

    float runmax[4] = {0.f, 0.f, 0.f, 0.f};   // relu output is >= 0
    const int abase0 = m16 + 2 * khalf;
    for (int i = 0; i < (LENG / 16) / 8; ++i) {          // 16 uniform iterations
        const int base = (i * 8 + wave) * 16 + abase0;   // xs[base] = xw[l0+m-1+k0]
        v2f a;
        a.x = xs[base];
        float ay = xs[base + 1];                         // in bounds due to tail pad
        a.y = (khalf == 0) ? ay : 0.f;                   // K=3 tap is zero (cndmask)
        #pragma unroll
        for (int nt = 0; nt < 4; ++nt) {
            v8f d = __builtin_amdgcn_wmma_f32_16x16x4_f32(
                false, a, false, bmat[nt], (short)0, cbias[nt], false, false);
            float mx = runmax[nt];
            #pragma unroll
            for (int r = 0; r < 8; ++r) mx = fmaxf(mx, fmaxf(d[r], 0.f));  // max3
            runmax[nt] = mx;
        }
    }
    // lanes t and t^16 hold the same filter column (rows 0-7 vs 8-15)
    #pragma unroll
    for (int nt = 0; nt < 4; ++nt) {
        float m2 = fmaxf(runmax[nt], __shfl_xor(runmax[nt], 16, 32));
        if (lane < 16) wavemax[wave][nt * 16 + m16] = m2;
    }
    __syncthreads();

    float* frow = fused + (size_t)b * CLSIN + s * (NFIL + 4);
    if (tid < NFIL) {
        float m = wavemax[0][tid];
        #pragma unroll
        for (int w = 1; w < 8; ++w) m = fmaxf(m, wavemax[w][tid]);
        frow[tid] = m;
    }
    if (tid == 0) {
        float4 r = red[0];
        float mean = r.x * (1.f / LENG);
        float var  = r.y * (1.f / LENG) - mean * mean;
        frow[NFIL + 0] = mean;
        frow[NFIL + 1] = var;
        frow[NFIL + 2] = r.z;
        frow[NFIL + 3] = r.w;
    }
}

// ---------------- MLP: layer1 via WMMA f32, layers 2-4 scalar ----------------
__global__ __launch_bounds__(256) void mlp_kernel(const float* __restrict__ fused,
                                                  const float* __restrict__ w1, const float* __restrict__ b1,
                                                  const float* __restrict__ w2, const float* __restrict__ b2,
                                                  const float* __restrict__ w3, const float* __restrict__ b3,
                                                  const float* __restrict__ w4, const float* __restrict__ b4,
                                                  float* __restrict__ out) {
    __shared__ float h1[NBAT][64];
    __shared__ float h2[NBAT][32];
    __shared__ float h3[NBAT][16];

    const int tid   = threadIdx.x;
    const int wave  = __builtin_amdgcn_readfirstlane(tid >> 5);
    const int lane  = tid & 31;
    const int m16   = lane & 15;
    const int khalf = lane >> 4;
    const int mtile = wave & 1;    // 2 M-tiles (32 rows)
    const int ntile = wave >> 1;   // 4 N-tiles (64 cols)
    const int n     = ntile * 16 + m16;

    // layer 1: [32 x 2176] @ [2176 x 64], K swept 4 at a time, bias pre-loaded in C
    v8f acc;
    {
        float bn = b1[n];
        #pragma unroll
        for (int r = 0; r < 8; ++r) acc[r] = bn;
    }
    const float* arow = fused + (size_t)(mtile * 16 + m16) * CLSIN;
    for (int ks = 0; ks < CLSIN / 4; ++ks) {
        const int k0 = ks * 4 + 2 * khalf;
        v2f a;  a.x  = arow[k0];
                a.y  = arow[k0 + 1];
        v2f bb; bb.x = w1[(size_t)k0 * 64 + n];
                bb.y = w1[(size_t)(k0 + 1) * 64 + n];
        acc = __builtin_amdgcn_wmma_f32_16x16x4_f32(
            false, a, false, bb, (short)0, acc, false, false);
    }
    #pragma unroll
    for (int r = 0; r < 8; ++r) {
        int m = mtile * 16 + khalf * 8 + r;    // C/D row mapping
        h1[m][n] = fmaxf(acc[r], 0.f);
    }
    __syncthreads();

    // layer 2: 32x32, K=64
    for (int idx = tid; idx < NBAT * 32; idx += 256) {
        int m = idx >> 5, nn = idx & 31;
        float acc2 = b2[nn];
        #pragma unroll 8
        for (int k = 0; k < 64; ++k) acc2 += h1[m][k] * w2[k * 32 + nn];
        h2[m][nn] = fmaxf(acc2, 0.f);
    }
    __syncthreads();

    // layer 3: 32x16, K=32
    for (int idx = tid; idx < NBAT * 16; idx += 256) {
        int m = idx >> 4, nn = idx & 15;
        float acc3 = b3[nn];
        #pragma unroll 8
        for (int k = 0; k < 32; ++k) acc3 += h2[m][k] * w3[k * 16 + nn];
        h3[m][nn] = fmaxf(acc3, 0.f);
    }
    __syncthreads();

    // layer 4: 32x2, K=16, no relu
    if (tid < NBAT * 2) {
        int m = tid >> 1, nn = tid & 1;
        float acc4 = b4[nn];
        #pragma unroll
        for (int k = 0; k < 16; ++k) acc4 += h3[m][k] * w4[k * 2 + nn];
        out[m * 2 + nn] = acc4;
    }
}

extern "C" void kernel_launch(void* const* d_in, const int* in_sizes, int n_in,
                              void* d_out, int out_size, void* d_ws, size_t ws_size,
                              hipStream_t stream) {
    const float* x              = (const float*)d_in[0];
    const float* conv_w         = (const float*)d_in[1];
    const float* conv_b         = (const float*)d_in[2];
    const float* sensor_weights = (const float*)d_in[3];
    const float* w1 = (const float*)d_in[4];
    const float* b1 = (const float*)d_in[5];
    const float* w2 = (const float*)d_in[6];
    const float* b2 = (const float*)d_in[7];
    const float* w3 = (const float*)d_in[8];
    const float* b3 = (const float*)d_in[9];
    const float* w4 = (const float*)d_in[10];
    const float* b4 = (const float*)d_in[11];

    float* ws    = (float*)d_ws;
    float* sw    = ws;        // 32 floats (softmaxed sensor weights)
    float* fused = ws + 64;   // 32 * 2176 floats, MLP-ready layout

    softmax32_kernel<<<1, 32, 0, stream>>>(sensor_weights, sw);
    conv_kernel<<<NBAT * NSEN, 256, 0, stream>>>(x, conv_w, conv_b, sw, fused);
    mlp_kernel<<<1, 256, 0, stream>>>(fused, w1, b1, w2, b2, w3, b3, w4, b4,
                                      (float*)d_out);
}